// StabilizedVQ_23536420782588
// MI455X (gfx1250) — compile-verified
//
#include <hip/hip_runtime.h>
#include <hip/hip_bf16.h>
#include <math.h>

typedef __attribute__((ext_vector_type(2))) float v2f;
typedef __attribute__((ext_vector_type(8))) float v8f;

#define NTOK  16384
#define DIMC  256
#define DFULL 512
#define VOCAB 8192

// Dynamic LDS layout for vq_argmin (bytes):
//   float lds_z[16][514]        @ 0        (32896)
//   float lds_w[2][256][34]     @ 32896    (2 x 34816)
//   float red_v[4][16]          @ 102528   (256)
//   int   red_i[4][16]          @ 102784   (256)
#define LDSB_Z    0
#define LDSB_W    32896
#define LDSB_WBUF 34816
#define LDSB_RV   102528
#define LDSB_RI   102784
#define LDS_TOTAL 103040

// workspace layout (bytes):
//   int      idx[NTOK]        @ 0        (64 KB)
//   float    wsq[VOCAB]       @ 65536    (32 KB)
//   unsigned hist[VOCAB]      @ 98304    (32 KB)
//   float    loss_part[NTOK]  @ 131072   (64 KB)

__device__ __forceinline__ float wave_reduce_add(float v) {
#pragma unroll
  for (int m = 16; m >= 1; m >>= 1) v += __shfl_xor(v, m, 32);
  return v;
}

// Async-copy one codebook column (32 floats = 8 x b128) from global to LDS.
// GVS mode: saddr = uniform base, vaddr = per-lane byte offset; the 24-bit
// immediate offset is added to BOTH the LDS and global addresses (ISA §10.2),
// so consecutive 16B chunks advance both sides together. Tracked by ASYNCcnt.
__device__ __forceinline__ void async_copy_col(unsigned ldsOff, unsigned vOff,
                                               unsigned long long base) {
  asm volatile(
      "global_load_async_to_lds_b128 %0, %1, %2 offset:0\n\t"
      "global_load_async_to_lds_b128 %0, %1, %2 offset:16\n\t"
      "global_load_async_to_lds_b128 %0, %1, %2 offset:32\n\t"
      "global_load_async_to_lds_b128 %0, %1, %2 offset:48\n\t"
      "global_load_async_to_lds_b128 %0, %1, %2 offset:64\n\t"
      "global_load_async_to_lds_b128 %0, %1, %2 offset:80\n\t"
      "global_load_async_to_lds_b128 %0, %1, %2 offset:96\n\t"
      "global_load_async_to_lds_b128 %0, %1, %2 offset:112"
      :: "v"(ldsOff), "v"(vOff), "s"(base)
      : "memory");
}

__device__ __forceinline__ void wait_async0() {
  asm volatile("s_wait_asynccnt 0x0" ::: "memory");
}

// ---------------------------------------------------------------- kernel 0
// wsq[v] = sum_k W[v][k]^2  (one wave per codebook row), and zero histogram.
__global__ void __launch_bounds__(256) vq_prep(const float* __restrict__ w,
                                               float* __restrict__ wsq,
                                               unsigned* __restrict__ hist) {
  const int tid = threadIdx.x;
  const int gid = blockIdx.x * 256 + tid;
  if (gid < VOCAB) hist[gid] = 0u;

  const int wave = tid >> 5, lane = tid & 31;
  const int v = blockIdx.x * 8 + wave;
  const float* row = w + (size_t)v * DFULL;
  float s = 0.f;
#pragma unroll
  for (int j = 0; j < 16; ++j) {
    float x = row[lane + 32 * j];
    s += x * x;
  }
  s = wave_reduce_add(s);
  if (lane == 0) wsq[v] = s;
}

// ---------------------------------------------------------------- kernel 1
// Fused distance-GEMM + row argmin. 4 waves / 16 tokens per workgroup.
// Column chunks of 256; each wave owns 64 cols = 4 WMMA accumulator tiles.
// W tiles stream via double-buffered GLOBAL_LOAD_ASYNC_TO_LDS_B128.
__global__ void __launch_bounds__(128) vq_argmin(
    const float* __restrict__ zr, const float* __restrict__ zi,
    const float* __restrict__ w, const float* __restrict__ wsq,
    int* __restrict__ out_idx) {
  extern __shared__ char smem[];
  float* lds_z = (float*)(smem + LDSB_Z);    // [16][514]
  float* lds_w = (float*)(smem + LDSB_W);    // [2][256][34]
  float* red_v = (float*)(smem + LDSB_RV);   // [4][16]
  int*   red_i = (int*)(smem + LDSB_RI);     // [4][16]
  const unsigned ldsWBase = (unsigned)(uintptr_t)(smem + LDSB_W);

  const int tid    = threadIdx.x;
  const int lane   = tid & 31;
  const int waveId = tid >> 5;
  const int half   = lane >> 4;        // 0: lanes 0-15, 1: lanes 16-31
  const int ml     = lane & 15;        // A: row M; B: col N within 16-tile
  const int koff   = half * 2;         // upper half-wave holds K+2 (ISA §7.12.2)
  const int rowBase = blockIdx.x * 16;

  // ---- stage z tile once: z_flat[r][k] = k<256 ? z_real : z_imag
  for (int i = 0; i < 64; ++i) {
    int f = tid + 128 * i;
    int r = f >> 9;
    int k = f & 511;
    float v = (k < DIMC) ? zr[(size_t)(rowBase + r) * DIMC + k]
                         : zi[(size_t)(rowBase + r) * DIMC + (k - DIMC)];
    lds_z[r * 514 + k] = v;
  }

  float bestv[8];
  int   besti[8];
#pragma unroll
  for (int r = 0; r < 8; ++r) { bestv[r] = 3.4e38f; besti[r] = 0; }

  // stage s covers column chunk (s>>4)*256, K block (s&15)*32, buffer (s&1).
  const int NSTAGE = (VOCAB / 256) * 16;  // 512

  // prologue: issue stage 0 (2 columns per thread, 8 x b128 each)
  {
    const float* base = w;  // chunk 0, kb 0
    async_copy_col(ldsWBase + (unsigned)tid * 136u, (unsigned)tid * 2048u,
                   (unsigned long long)base);
    async_copy_col(ldsWBase + (unsigned)(tid + 128) * 136u,
                   (unsigned)(tid + 128) * 2048u, (unsigned long long)base);
  }

  v8f acc0 = {}, acc1 = {}, acc2 = {}, acc3 = {};

  for (int s = 0; s < NSTAGE; ++s) {
    wait_async0();     // this wave's stage-s async writes have landed in LDS
    __syncthreads();   // publish to all waves; also fences last reads of buf

    if (s + 1 < NSTAGE) {  // overlap next stage's copy with current compute
      const int s1 = s + 1;
      const float* base =
          w + (size_t)((s1 >> 4) * 256) * DFULL + (s1 & 15) * 32;
      const unsigned bufOff = ldsWBase + (unsigned)(s1 & 1) * LDSB_WBUF;
      async_copy_col(bufOff + (unsigned)tid * 136u, (unsigned)tid * 2048u,
                     (unsigned long long)base);
      async_copy_col(bufOff + (unsigned)(tid + 128) * 136u,
                     (unsigned)(tid + 128) * 2048u, (unsigned long long)base);
    }

    if ((s & 15) == 0) {
      acc0 = (v8f){}; acc1 = (v8f){}; acc2 = (v8f){}; acc3 = (v8f){};
    }

    const float* zrow = lds_z + ml * 514 + (s & 15) * 32 + koff;
    const float* wbuf =
        lds_w + (s & 1) * (256 * 34) + (waveId * 64 + ml) * 34 + koff;
#pragma unroll
    for (int kk = 0; kk < 32; kk += 4) {
      v2f a  = *(const v2f*)(zrow + kk);
      v2f b0 = *(const v2f*)(wbuf + 0 * 544 + kk);   // 544 = 16 cols * 34
      v2f b1 = *(const v2f*)(wbuf + 1 * 544 + kk);
      v2f b2 = *(const v2f*)(wbuf + 2 * 544 + kk);
      v2f b3 = *(const v2f*)(wbuf + 3 * 544 + kk);
      acc0 = __builtin_amdgcn_wmma_f32_16x16x4_f32(
          false, a, false, b0, (short)0, acc0, false, false);
      acc1 = __builtin_amdgcn_wmma_f32_16x16x4_f32(
          false, a, false, b1, (short)0, acc1, false, false);
      acc2 = __builtin_amdgcn_wmma_f32_16x16x4_f32(
          false, a, false, b2, (short)0, acc2, false, false);
      acc3 = __builtin_amdgcn_wmma_f32_16x16x4_f32(
          false, a, false, b3, (short)0, acc3, false, false);
    }

    if ((s & 15) == 15) {  // epilogue: score = ||w||^2 - 2 z.w ; fold argmin
      const int c0 = (s >> 4) * 256 + waveId * 64 + ml;
      const float w0 = wsq[c0];
      const float w1 = wsq[c0 + 16];
      const float w2 = wsq[c0 + 32];
      const float w3 = wsq[c0 + 48];
#pragma unroll
      for (int r = 0; r < 8; ++r) {
        float s0 = fmaf(-2.0f, acc0[r], w0);
        float s1 = fmaf(-2.0f, acc1[r], w1);
        float s2 = fmaf(-2.0f, acc2[r], w2);
        float s3 = fmaf(-2.0f, acc3[r], w3);
        if (s0 < bestv[r]) { bestv[r] = s0; besti[r] = c0; }
        if (s1 < bestv[r]) { bestv[r] = s1; besti[r] = c0 + 16; }
        if (s2 < bestv[r]) { bestv[r] = s2; besti[r] = c0 + 32; }
        if (s3 < bestv[r]) { bestv[r] = s3; besti[r] = c0 + 48; }
      }
    }
  }

  // cross-lane argmin within each 16-lane half (rows r and r+8)
#pragma unroll
  for (int m = 8; m >= 1; m >>= 1) {
#pragma unroll
    for (int r = 0; r < 8; ++r) {
      float ov = __shfl_xor(bestv[r], m, 32);
      int   oi = __shfl_xor(besti[r], m, 32);
      if (ov < bestv[r] || (ov == bestv[r] && oi < besti[r])) {
        bestv[r] = ov; besti[r] = oi;
      }
    }
  }
  if (ml == 0) {
#pragma unroll
    for (int r = 0; r < 8; ++r) {
      red_v[waveId * 16 + r + 8 * half] = bestv[r];
      red_i[waveId * 16 + r + 8 * half] = besti[r];
    }
  }
  __syncthreads();
  if (tid < 16) {  // combine the 4 waves' column slices, lowest-index ties
    float bv = red_v[0 * 16 + tid];
    int   bi = red_i[0 * 16 + tid];
#pragma unroll
    for (int wv = 1; wv < 4; ++wv) {
      float ov = red_v[wv * 16 + tid];
      int   oi = red_i[wv * 16 + tid];
      if (ov < bv || (ov == bv && oi < bi)) { bv = ov; bi = oi; }
    }
    out_idx[rowBase + tid] = bi;
  }
}

// ---------------------------------------------------------------- kernel 2
// Gather W[idx], emit interleaved complex64, per-token loss partial, histogram.
__global__ void __launch_bounds__(256) vq_gather(
    const float* __restrict__ zr, const float* __restrict__ zi,
    const float* __restrict__ w, const int* __restrict__ idx,
    float* __restrict__ out, float* __restrict__ loss_part,
    unsigned* __restrict__ hist) {
  const int wave = threadIdx.x >> 5, lane = threadIdx.x & 31;
  const int n = blockIdx.x * 8 + wave;
  const int code = idx[n];
  const float* wrow = w + (size_t)code * DFULL;
  float2* out2 = (float2*)out + (size_t)n * DIMC;
  float loss = 0.f;
#pragma unroll
  for (int j = 0; j < 8; ++j) {
    int d = lane + 32 * j;
    float wr = wrow[d];
    float wi = wrow[DIMC + d];
    float a = zr[(size_t)n * DIMC + d];
    float b = zi[(size_t)n * DIMC + d];
    out2[d] = make_float2(wr, wi);  // complex64 interleave
    float dr = wr - a, di = wi - b;
    loss += dr * dr + di * di;
  }
  loss = wave_reduce_add(loss);
  if (lane == 0) {
    loss_part[n] = loss;
    atomicAdd(&hist[code], 1u);     // integer atomics: deterministic
  }
}

// ---------------------------------------------------------------- kernel 3
// Deterministic fixed-order reductions -> vq_loss, perplexity.
__global__ void __launch_bounds__(256) vq_finalize(
    const float* __restrict__ loss_part, const unsigned* __restrict__ hist,
    float* __restrict__ out) {
  __shared__ float s1[256];
  __shared__ float s2[256];
  const int t = threadIdx.x;
  float ls = 0.f;
  for (int j = 0; j < NTOK / 256; ++j) ls += loss_part[t + 256 * j];
  float es = 0.f;
  for (int j = 0; j < VOCAB / 256; ++j) {
    float p = (float)hist[t + 256 * j] * (1.0f / (float)NTOK);
    es += p * logf(p + 1e-10f);
  }
  s1[t] = ls;
  s2[t] = es;
  __syncthreads();
  for (int off = 128; off >= 1; off >>= 1) {
    if (t < off) { s1[t] += s1[t + off]; s2[t] += s2[t + off]; }
    __syncthreads();
  }
  if (t == 0) {
    // codebook + 0.25*commitment == 1.25 * mean((z_q - z)^2)
    out[(size_t)NTOK * DFULL + 0] = 1.25f * s1[0] / (float)((size_t)NTOK * DFULL);
    out[(size_t)NTOK * DFULL + 1] = expf(-s2[0]);
  }
}

// ---------------------------------------------------------------- launch
extern "C" void kernel_launch(void* const* d_in, const int* in_sizes, int n_in,
                              void* d_out, int out_size, void* d_ws, size_t ws_size,
                              hipStream_t stream) {
  const float* zr = (const float*)d_in[0];
  const float* zi = (const float*)d_in[1];
  const float* w  = (const float*)d_in[2];
  float* out = (float*)d_out;

  char* ws = (char*)d_ws;
  int*      idx  = (int*)(ws + 0);
  float*    wsq  = (float*)(ws + 65536);
  unsigned* hist = (unsigned*)(ws + 98304);
  float*    lp   = (float*)(ws + 131072);

  vq_prep    <<<VOCAB / 8, 256, 0, stream>>>(w, wsq, hist);
  vq_argmin  <<<NTOK / 16, 128, LDS_TOTAL, stream>>>(zr, zi, w, wsq, idx);
  vq_gather  <<<NTOK / 8, 256, 0, stream>>>(zr, zi, w, idx, out, lp, hist);
  vq_finalize<<<1, 256, 0, stream>>>(lp, hist, out);
}